// MeshTokenizer_32976758899158
// MI455X (gfx1250) — compile-verified
//
#include <hip/hip_runtime.h>
#include <stdint.h>

// ---------------------------------------------------------------------------
// MeshTokenizer for MI455X (gfx1250, wave32).
//
// Pipeline:
//  K0 init       : zero global-max accumulator, write -1 borders.
//  K1 minmax     : per-(vertex,coord) min/max over batch dim -> center;
//                  global max range ("longest") via bit-cast atomicMax.
//  K2 normalize  : v = (vertices - center) / longest.
//  Vertex sort   : per-batch stable lexsort by (z,y,x): 3 stages of LSD radix
//                  (8-bit digits, 4 passes each) on sortable float keys,
//                  payload = vertex index.
//  Face sort     : per-batch key-only sort of 45-bit packed (z<<30|y<<15|x)
//                  reindexed ids, 6 byte-passes.
//  Tokenize      : TDM-stage face chunks to LDS, gather sorted_v, quantize
//                  (rintf = round-half-even), emit ids/attn/codes/dq outputs.
//
// Occupancy design: each radix pass is decomposed chip-wide as
//   hist (32x32 WGs) -> scan (32 WGs) -> scatter (32x32 WGs),
// with kernel boundaries as the cross-WG sync. Ordering unit = wave-segment
// (NSEG=256 segments per batch); counters cnt[b][digit][seg] are scanned
// digit-major/segment-minor per batch, intra-wave stability via wave32
// ballot peer-masks. All ping-pong buffers (~68 MB ws) are L2-resident
// (192 MB L2). Every hist/scatter WG stages its contiguous record chunk into
// LDS with one tensor_load_to_lds (TDM) synced by s_wait_tensorcnt.
// ---------------------------------------------------------------------------

#define PAD_ID   (-1)
#define BATCH    32
#define NVERT    32768
#define NFACE    65536
#define RADIX    256
#define CHUNK    2048
#define SEQLEN   ((size_t)NFACE * 9 + 2)

#define SORT_WGS     32   // workgroups per batch per radix pass
#define SORT_THREADS 256  // 8 waves of 32
#define WPB          8    // waves per workgroup
#define NSEG         256  // wave-segments per batch (= SORT_WGS * WPB)

typedef unsigned long long u64;
typedef __attribute__((ext_vector_type(4))) unsigned v4u;
typedef __attribute__((ext_vector_type(8))) unsigned v8u;

__device__ __forceinline__ unsigned f2key(float f) {
  unsigned u = __float_as_uint(f);
  return (u & 0x80000000u) ? ~u : (u | 0x80000000u);  // monotone float->u32
}

__device__ __forceinline__ unsigned rfl(unsigned x) {
  return (unsigned)__builtin_amdgcn_readfirstlane((int)x);
}

// TDM: DMA ndw dwords (contiguous) from global to LDS. Caller issues from one
// wave and follows with s_wait_tensorcnt + workgroup barrier.
__device__ __forceinline__ void tdm_load(unsigned lds_addr, const void* gsrc,
                                         unsigned ndw) {
  u64 ga = (u64)(uintptr_t)gsrc;
  v4u g0;  // D# group 0: count=1 | lds_addr | global_addr[56:0] | type=2
  g0[0] = rfl(1u);
  g0[1] = rfl(lds_addr);
  g0[2] = rfl((unsigned)(ga & 0xFFFFFFFFull));
  g0[3] = rfl(((unsigned)((ga >> 32) & 0x1FFFFFFull)) | 0x80000000u);
  v8u g1;  // D# group 1: data_size=4B, 1D tile of ndw elems, stride0=ndw
  g1[0] = rfl(2u << 16);                              // data_size=2 (4B)
  g1[1] = rfl((ndw & 0xFFFFu) << 16);                 // tensor_dim0 lo16
  g1[2] = rfl(((ndw >> 16) & 0xFFFFu) | (1u << 16));  // dim0 hi16 | dim1=1
  g1[3] = rfl((ndw & 0xFFFFu) << 16);                 // tile_dim0 = ndw
  g1[4] = rfl(0u);                                    // tile_dim1/2 unused
  g1[5] = rfl(ndw);                                   // stride0 lo32
  g1[6] = rfl(0u);
  g1[7] = rfl(0u);
  asm volatile("tensor_load_to_lds %0, %1" ::"s"(g0), "s"(g1) : "memory");
}

// ---------------------------------------------------------------------------
__global__ void init_kernel(unsigned* maxbits, float* out_ids, float* out_attn) {
  int t = threadIdx.x;
  if (t == 0) *maxbits = 0u;
  if (t < BATCH) {
    out_ids[(size_t)t * SEQLEN] = -1.0f;
    out_ids[(size_t)t * SEQLEN + SEQLEN - 1] = -1.0f;
    out_attn[(size_t)t * SEQLEN] = -1.0f;
    out_attn[(size_t)t * SEQLEN + SEQLEN - 1] = -1.0f;
  }
}

__global__ __launch_bounds__(256) void minmax_kernel(const float* __restrict__ verts,
                                                     float* __restrict__ center,
                                                     unsigned* maxbits) {
  int j = blockIdx.x * 256 + threadIdx.x;  // over NVERT*3
  float mn = 0.f, mx = 0.f;
  if (j < NVERT * 3) {
    mn = mx = verts[j];
    for (int b = 1; b < BATCH; ++b) {
      float x = verts[(size_t)b * NVERT * 3 + j];
      mn = fminf(mn, x);
      mx = fmaxf(mx, x);
    }
    center[j] = 0.5f * (mn + mx);
  }
  float r = (j < NVERT * 3) ? (mx - mn) : 0.f;
  __shared__ float red[256];
  red[threadIdx.x] = r;
  __syncthreads();
  for (int s = 128; s > 0; s >>= 1) {
    if ((int)threadIdx.x < s) red[threadIdx.x] = fmaxf(red[threadIdx.x], red[threadIdx.x + s]);
    __syncthreads();
  }
  if (threadIdx.x == 0) atomicMax(maxbits, __float_as_uint(red[0]));  // r >= 0
}

__global__ __launch_bounds__(256) void normalize_kernel(const float* __restrict__ verts,
                                                        const float* __restrict__ center,
                                                        const unsigned* maxbits,
                                                        float* __restrict__ v) {
  int i = blockIdx.x * 256 + threadIdx.x;  // over BATCH*NVERT*3
  if (i >= BATCH * NVERT * 3) return;
  float longest = __uint_as_float(*maxbits);
  int j = i % (NVERT * 3);
  v[i] = (verts[i] - center[j]) / longest;
}

// ---------------------------------------------------------------------------
// Radix pass, step 1: per-wave-segment digit histograms.
// NPW = elements per wave-segment (128 vertices / 256 faces).
// ---------------------------------------------------------------------------
template <int NPW>
__global__ __launch_bounds__(SORT_THREADS) void hist_kernel(
    const u64* __restrict__ rec, unsigned* __restrict__ cnt_g, int shift) {
  __shared__ unsigned cl[WPB][RADIX];
  __shared__ u64 tile[NPW * WPB];
  const int b = blockIdx.y, wg = blockIdx.x;
  const int t = threadIdx.x, w = t >> 5, lane = t & 31;
  for (int i = t; i < WPB * RADIX; i += SORT_THREADS) (&cl[0][0])[i] = 0u;
  const u64* src = rec + (size_t)b * (NPW * NSEG) + (size_t)wg * (NPW * WPB);
  if (t < 32) {  // wave 0: one DMA for the whole WG chunk
    tdm_load((unsigned)(uintptr_t)(&tile[0]), src, (unsigned)(NPW * WPB * 2));
    __builtin_amdgcn_s_wait_tensorcnt(0);
  }
  __syncthreads();
  const u64* my = tile + w * NPW;
  for (int k = 0; k < NPW / 32; ++k) {
    unsigned d = (unsigned)(my[(k << 5) + lane] >> shift) & 0xFFu;
    atomicAdd(&cl[w][d], 1u);
  }
  __syncthreads();
  unsigned* cg = cnt_g + (size_t)b * (RADIX * NSEG);
  for (int i = t; i < WPB * RADIX; i += SORT_THREADS) {
    int w2 = i >> 8, d = i & 255;
    cg[d * NSEG + wg * WPB + w2] = cl[w2][d];
  }
}

// ---------------------------------------------------------------------------
// Radix pass, step 2: per-batch exclusive scan of 65536 counters
// (digit-major, segment-minor) -> stable global bases.
// ---------------------------------------------------------------------------
__global__ __launch_bounds__(1024) void scan_kernel(unsigned* __restrict__ cnt_g) {
  __shared__ unsigned wtot[32];
  const int b = blockIdx.x;
  const int t = threadIdx.x, w = t >> 5, lane = t & 31;
  unsigned* cg = cnt_g + (size_t)b * (RADIX * NSEG);
  unsigned sum = 0u;
  for (int k = 0; k < 64; ++k) sum += cg[t * 64 + k];
  unsigned incl = sum;
#pragma unroll
  for (int off = 1; off < 32; off <<= 1) {
    unsigned n = __shfl_up(incl, (unsigned)off, 32);
    if (lane >= off) incl += n;
  }
  if (lane == 31) wtot[w] = incl;
  unsigned exw = incl - sum;
  __syncthreads();
  if (w == 0) {
    unsigned q = wtot[lane], qi = q;
#pragma unroll
    for (int off = 1; off < 32; off <<= 1) {
      unsigned n = __shfl_up(qi, (unsigned)off, 32);
      if (lane >= off) qi += n;
    }
    wtot[lane] = qi - q;  // exclusive wave offsets
  }
  __syncthreads();
  unsigned run = wtot[w] + exw;
  for (int k = 0; k < 64; ++k) {
    unsigned x = cg[t * 64 + k];
    cg[t * 64 + k] = run;
    run += x;
  }
}

// ---------------------------------------------------------------------------
// Radix pass, step 3: stable scatter. Each wave consumes its segment in lane-
// batch order; ballot peer-masks give lane ranks; the peer leader bumps the
// (digit, segment) running offset (staged in LDS) with an LDS atomic.
// ---------------------------------------------------------------------------
template <int NPW>
__global__ __launch_bounds__(SORT_THREADS) void scatter_kernel(
    const u64* __restrict__ rec_in, u64* __restrict__ rec_out,
    const unsigned* __restrict__ cnt_g, int shift) {
  __shared__ unsigned cl[WPB][RADIX];
  __shared__ u64 tile[NPW * WPB];
  const int b = blockIdx.y, wg = blockIdx.x;
  const int t = threadIdx.x, w = t >> 5, lane = t & 31;
  const unsigned* cg = cnt_g + (size_t)b * (RADIX * NSEG);
  for (int i = t; i < WPB * RADIX; i += SORT_THREADS) {
    int w2 = i >> 8, d = i & 255;
    cl[w2][d] = cg[d * NSEG + wg * WPB + w2];
  }
  const u64* src = rec_in + (size_t)b * (NPW * NSEG) + (size_t)wg * (NPW * WPB);
  if (t < 32) {
    tdm_load((unsigned)(uintptr_t)(&tile[0]), src, (unsigned)(NPW * WPB * 2));
    __builtin_amdgcn_s_wait_tensorcnt(0);
  }
  __syncthreads();
  u64* outb = rec_out + (size_t)b * (NPW * NSEG);
  const u64* my = tile + w * NPW;
  for (int k = 0; k < NPW / 32; ++k) {
    u64 r = my[(k << 5) + lane];
    unsigned d = (unsigned)(r >> shift) & 0xFFu;
    unsigned m = 0xFFFFFFFFu;
#pragma unroll
    for (int bi = 0; bi < 8; ++bi) {
      unsigned bb = (unsigned)__ballot((int)((d >> bi) & 1u));
      m &= ((d >> bi) & 1u) ? bb : ~bb;
    }
    unsigned rank = (unsigned)__popc(m & ((1u << lane) - 1u));
    int leader = __ffs(m) - 1;
    unsigned bpos = 0u;
    if (lane == leader) bpos = atomicAdd(&cl[w][d], (unsigned)__popc(m));
    bpos = __shfl(bpos, leader, 32);
    outb[bpos + rank] = r;
  }
}

// ---------------------------------------------------------------------------
__global__ __launch_bounds__(256) void build_vkeys_kernel(const float* __restrict__ v,
                                                          u64* __restrict__ A, int c) {
  const int b = blockIdx.y;
  const int base = blockIdx.x * 1024;
  const float* vb = v + (size_t)b * NVERT * 3;
  u64* Ab = A + (size_t)b * NVERT;
  for (int q = 0; q < 4; ++q) {
    int i = base + q * 256 + (int)threadIdx.x;
    unsigned idx = (c == 0) ? (unsigned)i : (unsigned)(Ab[i] & 0xFFFFFFFFull);
    unsigned kb = f2key(vb[(size_t)idx * 3 + c]);
    Ab[i] = ((u64)kb << 32) | (u64)idx;
  }
}

__global__ __launch_bounds__(256) void extract_v_kernel(
    const u64* __restrict__ A, const float* __restrict__ v,
    unsigned* __restrict__ inv, float* __restrict__ sorted_v) {
  const int b = blockIdx.y;
  const int i = blockIdx.x * 256 + (int)threadIdx.x;
  unsigned pi = (unsigned)(A[(size_t)b * NVERT + i] & 0xFFFFFFFFull);
  inv[(size_t)b * NVERT + pi] = (unsigned)i;  // inverse permutation
  const float* vb = v + (size_t)b * NVERT * 3;
  float* svb = sorted_v + (size_t)b * NVERT * 3;
  svb[(size_t)i * 3 + 0] = vb[(size_t)pi * 3 + 0];
  svb[(size_t)i * 3 + 1] = vb[(size_t)pi * 3 + 1];
  svb[(size_t)i * 3 + 2] = vb[(size_t)pi * 3 + 2];
}

__global__ __launch_bounds__(256) void build_fkeys_kernel(
    const int* __restrict__ faces, const unsigned* __restrict__ inv,
    u64* __restrict__ A) {
  const int b = blockIdx.y;
  const int i = blockIdx.x * 256 + (int)threadIdx.x;
  const int* fb = faces + ((size_t)b * NFACE + (size_t)i) * 3;
  const unsigned* invb = inv + (size_t)b * NVERT;
  // jnp.mod(f, 32768) == f & 32767 for power-of-two NVERT (handles -1 pad).
  u64 x = invb[fb[0] & (NVERT - 1)];
  u64 y = invb[fb[1] & (NVERT - 1)];
  u64 z = invb[fb[2] & (NVERT - 1)];
  A[(size_t)b * NFACE + i] = x | (y << 15) | (z << 30);  // 45-bit key, z primary
}

__global__ __launch_bounds__(256) void extract_f_kernel(const u64* __restrict__ A,
                                                        float* __restrict__ out_sf) {
  const int b = blockIdx.y;
  const int i = blockIdx.x * 256 + (int)threadIdx.x;
  u64 k = A[(size_t)b * NFACE + i];
  size_t o = ((size_t)b * NFACE + (size_t)i) * 3;
  out_sf[o + 0] = (float)(unsigned)(k & 32767u);
  out_sf[o + 1] = (float)(unsigned)((k >> 15) & 32767u);
  out_sf[o + 2] = (float)(unsigned)((k >> 30) & 32767u);
}

// ---------------------------------------------------------------------------
// Tokenize: TDM-stage a 2048-face chunk (6144 dwords, contiguous) into LDS,
// then gather sorted_v and quantize.
// ---------------------------------------------------------------------------
__global__ __launch_bounds__(256) void tokenize_kernel(
    const int* __restrict__ faces, const float* __restrict__ sorted_v,
    float* __restrict__ out_ids, float* __restrict__ out_attn,
    float* __restrict__ out_codes, float* __restrict__ out_dq) {
  __shared__ int lf[CHUNK * 3];  // 24 KB
  const int b = blockIdx.y;
  const int ch = blockIdx.x;
  const int* src = faces + ((size_t)b * NFACE + (size_t)ch * CHUNK) * 3;

  if (threadIdx.x < 32) {
    tdm_load((unsigned)(uintptr_t)(&lf[0]), src, (unsigned)(CHUNK * 3));
    __builtin_amdgcn_s_wait_tensorcnt(0);
  }
  __syncthreads();

  const float* svb = sorted_v + (size_t)b * NVERT * 3;
  const size_t idsBase = (size_t)b * SEQLEN + 1;
  const size_t cBase = (size_t)b * NFACE * 9;

#pragma unroll
  for (int q = 0; q < CHUNK / 256; ++q) {
    int fi = (int)threadIdx.x + q * 256;
    size_t gf = (size_t)ch * CHUNK + (size_t)fi;
    int f0 = lf[fi * 3 + 0], f1 = lf[fi * 3 + 1], f2 = lf[fi * 3 + 2];
    bool msk = (f0 != PAD_ID) && (f1 != PAD_ID) && (f2 != PAD_ID);
    int fn[3];
    fn[0] = msk ? f0 : 0;
    fn[1] = msk ? f1 : 0;
    fn[2] = msk ? f2 : 0;
#pragma unroll
    for (int k = 0; k < 3; ++k) {
      const float* vc = svb + (size_t)fn[k] * 3;
#pragma unroll
      for (int c = 0; c < 3; ++c) {
        float tq = (vc[c] + 1.0f) * 64.0f - 0.5f;  // (x+1)*0.5*128 - 0.5
        int d = (int)rintf(tq);                    // round half-to-even
        d = d < 0 ? 0 : (d > 127 ? 127 : d);
        float dqf = (float)d;
        float code = msk ? dqf : -1.0f;
        size_t e = gf * 9 + (size_t)(k * 3 + c);
        out_dq[cBase + e] = dqf;
        out_codes[cBase + e] = code;
        out_ids[idsBase + e] = code;
        out_attn[idsBase + e] = msk ? 1.0f : 0.0f;
      }
    }
  }
}

// ---------------------------------------------------------------------------
extern "C" void kernel_launch(void* const* d_in, const int* in_sizes, int n_in,
                              void* d_out, int out_size, void* d_ws, size_t ws_size,
                              hipStream_t stream) {
  (void)in_sizes; (void)n_in; (void)out_size; (void)ws_size;
  const float* vertices = (const float*)d_in[0];
  const int* faces = (const int*)d_in[1];
  float* out = (float*)d_out;

  // Workspace carve-up (~68 MB total, fully L2-resident on 192 MB L2).
  char* ws = (char*)d_ws;
  unsigned* maxbits = (unsigned*)ws;                                  // 256 B
  float* center = (float*)(ws + 256);                                 // 384 KB
  float* v = (float*)(ws + 256 + 393216);                             // 12 MB
  float* sorted_v = v + (size_t)BATCH * NVERT * 3;                    // 12 MB
  unsigned* inv = (unsigned*)(sorted_v + (size_t)BATCH * NVERT * 3);  // 4 MB
  u64* big = (u64*)((char*)inv + (size_t)BATCH * NVERT * 4);          // 32 MB
  u64* recA = big;                          // vertex sort ping
  u64* recB = big + (size_t)BATCH * NVERT;  // vertex sort pong
  u64* keyA = big;                          // face sort reuses region
  u64* keyB = big + (size_t)BATCH * NFACE;
  unsigned* cnt_g = (unsigned*)((char*)big + (size_t)2 * BATCH * NFACE * 8);  // 8 MB

  float* out_ids = out;
  float* out_attn = out_ids + (size_t)BATCH * SEQLEN;
  float* out_codes = out_attn + (size_t)BATCH * SEQLEN;
  float* out_dq = out_codes + (size_t)BATCH * NFACE * 9;
  float* out_sf = out_dq + (size_t)BATCH * NFACE * 9;

  init_kernel<<<1, 64, 0, stream>>>(maxbits, out_ids, out_attn);
  minmax_kernel<<<(NVERT * 3 + 255) / 256, 256, 0, stream>>>(vertices, center, maxbits);
  normalize_kernel<<<(BATCH * NVERT * 3 + 255) / 256, 256, 0, stream>>>(vertices, center,
                                                                        maxbits, v);

  const dim3 gsort(SORT_WGS, BATCH);

  // Vertex lexsort: stages x, y, z (z = primary); 4 byte-passes per stage.
  for (int c = 0; c < 3; ++c) {
    build_vkeys_kernel<<<dim3(NVERT / 1024, BATCH), 256, 0, stream>>>(v, recA, c);
    u64 *in = recA, *outp = recB;
    for (int p = 0; p < 4; ++p) {
      int shift = 32 + 8 * p;
      hist_kernel<NVERT / NSEG><<<gsort, SORT_THREADS, 0, stream>>>(in, cnt_g, shift);
      scan_kernel<<<BATCH, 1024, 0, stream>>>(cnt_g);
      scatter_kernel<NVERT / NSEG><<<gsort, SORT_THREADS, 0, stream>>>(in, outp, cnt_g,
                                                                       shift);
      u64* tmp = in; in = outp; outp = tmp;
    }
    // 4 passes -> result back in recA
  }
  extract_v_kernel<<<dim3(NVERT / 256, BATCH), 256, 0, stream>>>(recA, v, inv, sorted_v);

  // Face sort: 45-bit packed key, 6 byte-passes (key-only).
  build_fkeys_kernel<<<dim3(NFACE / 256, BATCH), 256, 0, stream>>>(faces, inv, keyA);
  {
    u64 *in = keyA, *outp = keyB;
    for (int p = 0; p < 6; ++p) {
      int shift = 8 * p;
      hist_kernel<NFACE / NSEG><<<gsort, SORT_THREADS, 0, stream>>>(in, cnt_g, shift);
      scan_kernel<<<BATCH, 1024, 0, stream>>>(cnt_g);
      scatter_kernel<NFACE / NSEG><<<gsort, SORT_THREADS, 0, stream>>>(in, outp, cnt_g,
                                                                      shift);
      u64* tmp = in; in = outp; outp = tmp;
    }
    // 6 passes -> result back in keyA
  }
  extract_f_kernel<<<dim3(NFACE / 256, BATCH), 256, 0, stream>>>(keyA, out_sf);

  dim3 g5(NFACE / CHUNK, BATCH);
  tokenize_kernel<<<g5, 256, 0, stream>>>(faces, sorted_v, out_ids, out_attn, out_codes,
                                          out_dq);
}